// InterpretableMultiHeadAttention_91336774517398
// MI455X (gfx1250) — compile-verified
//
#include <hip/hip_runtime.h>
#include <hip/hip_bf16.h>
#include <math.h>

// ---------------- problem constants (from reference) ----------------
constexpr int Bn = 4, SQ = 512, SK = 512, E = 256, H = 4, HD = 64;
constexpr float INV_SCALE = 0.125f;            // 1/sqrt(HD)
constexpr int M_ROWS = Bn * SQ;                // 2048

// ---------------- CDNA5 vector types for WMMA ----------------
typedef __attribute__((ext_vector_type(2))) float v2f;
typedef __attribute__((ext_vector_type(8))) float v8f;

// fp32 WMMA 16x16x4: D = A(16x4) * B(4x16) + C(16x16)
__device__ __forceinline__ v8f wmma_f32(v2f a, v2f b, v8f c) {
    return __builtin_amdgcn_wmma_f32_16x16x4_f32(
        /*neg_a=*/false, a, /*neg_b=*/false, b,
        /*c_mod=*/(short)0, c, /*reuse_a=*/false, /*reuse_b=*/false);
}

__device__ __forceinline__ float fast_tanh(float x) {
#if __has_builtin(__builtin_amdgcn_tanhf)
    return __builtin_amdgcn_tanhf(x);          // v_tanh_f32 (gfx1250 TRANS op)
#elif __has_builtin(__builtin_amdgcn_tanh_f32)
    return __builtin_amdgcn_tanh_f32(x);
#else
    return tanhf(x);
#endif
}

// =====================================================================
// Shared GEMM body: Y[.. x Ndim] tile = X[.. x Kdim] * W^T + bias
// (W is [N x K] row-major). One wave -> 16 rows x 32 cols (2 WMMA tiles).
// K-loop batches 4 k-steps of fragment loads (12 x b64) into registers
// BEFORE issuing the 8 WMMAs, so s_wait_loadcnt becomes progressive and
// the matrix pipe overlaps the loads instead of stalling per-op.
// Fragment layouts per CDNA5 ISA 7.12.2:
//   A 16x4 f32 : M = lane%16, K = r + 2*(lane/16)   -> float2 load
//   B 4x16 f32 : K = r + 2*(lane/16), N = lane%16   -> float2 load (W rows)
//   C 16x16    : M = r + 8*(lane/16), N = lane%16
// =====================================================================
__device__ __forceinline__ void gemm_xwT_body(
    const float* __restrict__ X, const float* __restrict__ W,
    const float* __restrict__ bias, float* __restrict__ Y,
    int Kdim, int Ndim, int row0, int col0) {
    const int lane = threadIdx.x & 31;
    const int lm = lane & 15;        // M / N index within tile
    const int ld = lane >> 4;        // 0/1 -> K-half select

    const float* Arow = X + (size_t)(row0 + lm) * Kdim + 2 * ld;
    const float* W0 = W + (size_t)(col0 + lm) * Kdim + 2 * ld;
    const float* W1 = W + (size_t)(col0 + 16 + lm) * Kdim + 2 * ld;

    v8f acc0 = {}, acc1 = {};
    for (int kk = 0; kk < Kdim; kk += 16) {
        v2f a[4], b0[4], b1[4];
#pragma unroll
        for (int u = 0; u < 4; ++u) {          // issue all 12 loads first
            const int ko = kk + 4 * u;
            a[u]  = *(const v2f*)(Arow + ko);
            b0[u] = *(const v2f*)(W0 + ko);
            b1[u] = *(const v2f*)(W1 + ko);
        }
#pragma unroll
        for (int u = 0; u < 4; ++u) {          // then 8 back-to-back WMMAs
            acc0 = wmma_f32(a[u], b0[u], acc0);
            acc1 = wmma_f32(a[u], b1[u], acc1);
        }
    }
#pragma unroll
    for (int r = 0; r < 8; ++r) {
        const int m = row0 + r + 8 * ld;
        Y[(size_t)m * Ndim + col0 + lm]      = acc0[r] + bias[col0 + lm];
        Y[(size_t)m * Ndim + col0 + 16 + lm] = acc1[r] + bias[col0 + 16 + lm];
    }
}

// Kernel 1/4: fused Q/K/V projections (blockIdx.z selects matrix).
// Block = 8 waves -> 128 rows x 32 cols. Grid (16, 8, 3) = 384 blocks.
__global__ __launch_bounds__(256)
void k_proj3(const float* __restrict__ q, const float* __restrict__ k,
             const float* __restrict__ v,
             const float* __restrict__ wq, const float* __restrict__ wk,
             const float* __restrict__ wv,
             const float* __restrict__ bq, const float* __restrict__ bk,
             const float* __restrict__ bv,
             float* __restrict__ Qp, float* __restrict__ Kp,
             float* __restrict__ Vp) {
    const float *X, *W, *bias;
    float* Y;
    if (blockIdx.z == 0)      { X = q; W = wq; bias = bq; Y = Qp; }
    else if (blockIdx.z == 1) { X = k; W = wk; bias = bk; Y = Kp; }
    else                      { X = v; W = wv; bias = bv; Y = Vp; }
    const int row0 = blockIdx.x * 128 + (threadIdx.x >> 5) * 16;
    const int col0 = blockIdx.y * 32;
    gemm_xwT_body(X, W, bias, Y, E, E, row0, col0);
}

// Kernel 4/4: output projection.
__global__ __launch_bounds__(256)
void k_outproj(const float* __restrict__ att, const float* __restrict__ wo,
               const float* __restrict__ bo, float* __restrict__ out) {
    const int row0 = blockIdx.x * 128 + (threadIdx.x >> 5) * 16;
    const int col0 = blockIdx.y * 32;
    gemm_xwT_body(att, wo, bo, out, E, E, row0, col0);
}

// =====================================================================
// Kernel 2/4: additive scores + softmax -> weights[b,h,q,k]
// One block handles (b,h) and 8 consecutive q rows.
// K rows staged in LDS in 64-row chunks, padded stride 65 (64-bank LDS ->
// conflict-free (kl+d) bank pattern). 1024 hw-tanh per thread.
// Softmax: one wave32 per q row via shfl_xor reductions.
// =====================================================================
__global__ __launch_bounds__(256)
void k_scores(const float* __restrict__ Qp, const float* __restrict__ Kp,
              const float* __restrict__ vvec, float* __restrict__ weights) {
    constexpr int QT = 8;
    const int qt = blockIdx.x % (SQ / QT);
    const int bh = blockIdx.x / (SQ / QT);
    const int b = bh / H, h = bh % H;
    const int q0 = qt * QT;
    const int tid = threadIdx.x;

    __shared__ float kbuf[64][65];     // padded: bank = (kl + d) & 63
    __shared__ float qrow[QT][64];
    __shared__ float vh[64];
    __shared__ float sc[QT][SK];

    if (tid < 64) vh[tid] = vvec[h * 64 + tid];
    for (int i = tid; i < QT * 64; i += 256) {
        const int qq = i >> 6, d = i & 63;
        qrow[qq][d] = Qp[(size_t)b * SQ * E + (size_t)(q0 + qq) * E + h * 64 + d];
    }
    __syncthreads();

    const float* Kbase = Kp + (size_t)b * SQ * E + h * 64;
    for (int c = 0; c < SK / 64; ++c) {
        const int k0 = c * 64;
        for (int i = tid; i < 64 * 64; i += 256) {
            const int kk = i >> 6, d = i & 63;
            kbuf[kk][d] = Kbase[(size_t)(k0 + kk) * E + d];
        }
        if (c + 1 < SK / 64) {   // global_prefetch_b8 of the next chunk
            const int kk = tid >> 2, part = tid & 3;
            __builtin_prefetch(Kbase + (size_t)(k0 + 64 + kk) * E + part * 16, 0, 1);
        }
        __syncthreads();
#pragma unroll
        for (int p = 0; p < 2; ++p) {
            const int id = tid + 256 * p;
            const int ql = id >> 6, kl = id & 63;
            const float* qr = qrow[ql];
            const float* kr = kbuf[kl];
            float s = 0.f;
#pragma unroll
            for (int d = 0; d < 64; ++d)
                s += vh[d] * fast_tanh(qr[d] + kr[d]);
            sc[ql][k0 + kl] = s * INV_SCALE;
        }
        __syncthreads();
    }

    // softmax: wave w owns q row w (8 waves, 8 rows), 16 values per lane
    const int wave = tid >> 5, lane = tid & 31;
    float vals[16];
    float mx = -INFINITY;
#pragma unroll
    for (int j = 0; j < 16; ++j) { vals[j] = sc[wave][lane + 32 * j]; mx = fmaxf(mx, vals[j]); }
#pragma unroll
    for (int off = 16; off; off >>= 1) mx = fmaxf(mx, __shfl_xor(mx, off));
    float sum = 0.f;
#pragma unroll
    for (int j = 0; j < 16; ++j) { vals[j] = __expf(vals[j] - mx); sum += vals[j]; }
#pragma unroll
    for (int off = 16; off; off >>= 1) sum += __shfl_xor(sum, off);
    const float inv = 1.0f / sum;
    float* wout = weights + ((size_t)bh * SQ + q0 + wave) * SK;
#pragma unroll
    for (int j = 0; j < 16; ++j) wout[lane + 32 * j] = vals[j] * inv;
}

// =====================================================================
// Kernel 3/4: attended[b,q,h*64+d] = sum_k weights[b,h,q,k] * Vp[b,k,h*64+d]
// Per (b,h): GEMM M=512(q), N=64(d), K=512(k), fp32 WMMA.
// Block = 8 waves -> 64 rows x 64 cols (wave: 16 rows x 32 cols).
// Same batched-load scheme: 4 a-loads + 16 b-loads, then 8 WMMAs.
// =====================================================================
__global__ __launch_bounds__(256)
void k_attend(const float* __restrict__ Wt, const float* __restrict__ Vp,
              float* __restrict__ att) {
    const int bh = blockIdx.y;
    const int b = bh / H, h = bh % H;
    const int wave = threadIdx.x >> 5;
    const int lane = threadIdx.x & 31;
    const int lm = lane & 15, ld = lane >> 4;
    const int q0 = blockIdx.x * 64 + (wave >> 1) * 16;
    const int col0 = (wave & 1) * 32;

    const float* Arow = Wt + (size_t)bh * SQ * SK + (size_t)(q0 + lm) * SK + 2 * ld;
    const float* Bm = Vp + (size_t)b * SQ * E + h * 64;

    v8f acc0 = {}, acc1 = {};
    for (int kk = 0; kk < SK; kk += 16) {
        v2f a[4], b0[4], b1[4];
#pragma unroll
        for (int u = 0; u < 4; ++u) {          // batch all loads first
            const int ko = kk + 4 * u + 2 * ld;
            a[u] = *(const v2f*)(Arow + kk + 4 * u);
            b0[u].x = Bm[(size_t)ko * E + col0 + lm];
            b0[u].y = Bm[(size_t)(ko + 1) * E + col0 + lm];
            b1[u].x = Bm[(size_t)ko * E + col0 + 16 + lm];
            b1[u].y = Bm[(size_t)(ko + 1) * E + col0 + 16 + lm];
        }
#pragma unroll
        for (int u = 0; u < 4; ++u) {
            acc0 = wmma_f32(a[u], b0[u], acc0);
            acc1 = wmma_f32(a[u], b1[u], acc1);
        }
    }
#pragma unroll
    for (int r = 0; r < 8; ++r) {
        const int qrow = q0 + r + 8 * ld;
        float* dst = att + (size_t)b * SQ * E + (size_t)qrow * E + h * 64;
        dst[col0 + lm]      = acc0[r];
        dst[col0 + 16 + lm] = acc1[r];
    }
}

// =====================================================================
// host-side launcher
// =====================================================================
extern "C" void kernel_launch(void* const* d_in, const int* in_sizes, int n_in,
                              void* d_out, int out_size, void* d_ws, size_t ws_size,
                              hipStream_t stream) {
    const float* query = (const float*)d_in[0];
    const float* key   = (const float*)d_in[1];
    const float* value = (const float*)d_in[2];
    const float* wq_w  = (const float*)d_in[3];
    const float* wq_b  = (const float*)d_in[4];
    const float* wk_w  = (const float*)d_in[5];
    const float* wk_b  = (const float*)d_in[6];
    const float* wv_w  = (const float*)d_in[7];
    const float* wv_b  = (const float*)d_in[8];
    const float* vvec  = (const float*)d_in[9];
    const float* wo_w  = (const float*)d_in[10];
    const float* wo_b  = (const float*)d_in[11];
    (void)in_sizes; (void)n_in; (void)out_size; (void)ws_size;

    // workspace layout (floats): Qp | Kp | Vp | att  (4 x 2 MB = 8 MB)
    float* ws  = (float*)d_ws;
    const size_t MAT = (size_t)M_ROWS * E;   // 524288
    float* Qp  = ws;
    float* Kp  = ws + MAT;
    float* Vp  = ws + 2 * MAT;
    float* att = ws + 3 * MAT;

    // d_out: output [B,SQ,E] then weights [B,H,SQ,SK]
    float* outF    = (float*)d_out;
    float* weights = outF + MAT;

    const dim3 blk(256);

    // fused Q/K/V projections: grid (rows/128, cols/32, 3)
    k_proj3<<<dim3(M_ROWS / 128, E / 32, 3), blk, 0, stream>>>(
        query, key, value, wq_w, wk_w, wv_w, wq_b, wk_b, wv_b, Qp, Kp, Vp);

    // additive scores + softmax (dominant tanh work)
    k_scores<<<dim3(Bn * H * (SQ / 8)), blk, 0, stream>>>(Qp, Kp, vvec, weights);

    // attended = weights @ V_heads
    k_attend<<<dim3(SQ / 64, Bn * H), blk, 0, stream>>>(weights, Vp, att);

    // output projection
    k_outproj<<<dim3(M_ROWS / 128, E / 32), blk, 0, stream>>>(att, wo_w, wo_b, outF);
}